// ContextAwareHead_52527450030157
// MI455X (gfx1250) — compile-verified
//
#include <hip/hip_runtime.h>
#include <hip/hip_bf16.h>
#include <cfloat>

// MI455X / gfx1250: wave32, WMMA f32 16x16x4 used for all heavy GEMMs
// (K-projection layers + QK^T scores) to keep fp32 accuracy vs. the fp32
// reference while getting matrix-unit throughput (~8x VALU f32).
// Weight tiles stored row-pair-interleaved in LDS so every WMMA B fragment
// is a single aligned ds_load_b64 (no register repacking movs).

typedef float v2f __attribute__((ext_vector_type(2)));
typedef float v8f __attribute__((ext_vector_type(8)));

#define NB    8192
#define NRSS  12
#define NLED  256
#define NHID  128
#define NHD   128
#define CW_OFF ((size_t)NB * NRSS * NLED)

__device__ __forceinline__ v8f wmma4(v2f a, v2f b, v8f c) {
  // D = A(16x4,f32) x B(4x16,f32) + C(16x16,f32)
  return __builtin_amdgcn_wmma_f32_16x16x4_f32(false, a, false, b, (short)0, c,
                                               false, false);
}
__device__ __forceinline__ v8f splat8(float x) {
  v8f v;
#pragma unroll
  for (int i = 0; i < 8; ++i) v[i] = x;
  return v;
}
// Pair-interleaved B fragment: rows (row0,row0+1), row0 even, stored
// contiguously at [row0/2][col*2] -> one aligned b64 LDS load.
__device__ __forceinline__ v2f ldsBp(const float* base, int pstride, int row0,
                                     int colx) {
  const float* p = base + (row0 >> 1) * pstride + (colx << 1);
  v2f r;
  r.x = p[0];
  r.y = p[1];
  return r;
}

// ---------------- Kernel 1: motion encoder, channel weights, sigma ----------
__global__ __launch_bounds__(256) void k1_motion(
    const float* __restrict__ lstm, const float* __restrict__ w1,
    const float* __restrict__ b1, const float* __restrict__ w2,
    const float* __restrict__ b2, const float* __restrict__ ciw,
    const float* __restrict__ cib, const float* __restrict__ ssw,
    const float* __restrict__ ssb, float* __restrict__ out,
    float* __restrict__ mc_ws, float* __restrict__ sig_ws) {
  __shared__ float ls[16 * 128];
  __shared__ float h1[16 * 64];
  __shared__ float mcl[16 * 32];
  const int t = threadIdx.x;
  const int bt = blockIdx.x * 16;

  for (int i = t; i < 16 * 128; i += 256) ls[i] = lstm[(size_t)bt * 128 + i];
  __syncthreads();

#pragma unroll
  for (int i = 0; i < 4; ++i) {
    int e = t + 256 * i, r = e >> 6, c = e & 63;
    float acc = b1[c];
    for (int k = 0; k < 128; ++k) acc = fmaf(ls[r * 128 + k], w1[k * 64 + c], acc);
    h1[r * 64 + c] = fmaxf(acc, 0.f);
  }
  __syncthreads();

#pragma unroll
  for (int i = 0; i < 2; ++i) {
    int e = t + 256 * i, r = e >> 5, c = e & 31;
    float acc = b2[c];
    for (int k = 0; k < 64; ++k) acc = fmaf(h1[r * 64 + k], w2[k * 32 + c], acc);
    mcl[r * 32 + c] = acc;
    mc_ws[(size_t)(bt + r) * 32 + c] = acc;
  }
  __syncthreads();

  if (t < 192) {
    int r = t / 12, q = t % 12;
    float acc = cib[q];
    for (int k = 0; k < 32; ++k) acc = fmaf(mcl[r * 32 + k], ciw[k * 12 + q], acc);
    out[CW_OFF + (size_t)(bt + r) * 12 + q] = 1.f / (1.f + expf(-acc));
  }
  if (t < 16) {
    float acc = ssb[0];
    for (int k = 0; k < 32; ++k) acc = fmaf(mcl[t * 32 + k], ssw[k], acc);
    float sp = (acc > 20.f) ? acc : log1pf(expf(acc));
    sig_ws[bt + t] = sp + 0.5f;
  }
}

// ---------------- Kernel 2: fused K-projection + scores + softmax -----------
// One block per batch element, 256 threads = 8 waves. Each wave owns two
// 16-LED tiles and pushes them through K1(WMMA) -> K2(WMMA) -> scores^T(WMMA).
__global__ __launch_bounds__(256) void k2_attn(
    const float* __restrict__ rss_t, const float* __restrict__ ledf,
    const float* __restrict__ ledp, const float* __restrict__ prevp,
    const int* __restrict__ fmask, const float* __restrict__ qw1,
    const float* __restrict__ qb1, const float* __restrict__ qw2,
    const float* __restrict__ qb2, const float* __restrict__ kw1,
    const float* __restrict__ kb1, const float* __restrict__ kw2,
    const float* __restrict__ kb2, const float* __restrict__ mc_ws,
    const float* __restrict__ sig_ws, float* __restrict__ out) {
  extern __shared__ float sm[];
  float* w1p  = sm;                  // [6][96]   kp_w1 pair-interleaved
  float* w2p  = w1p + 6 * 96;        // [16][288] kp_w2 pair-interleaved
  float* Qs   = w2p + 16 * 288;      // [16][132] Q * cw
  float* qi   = Qs + 16 * 132;       // [16][36]  query input (rss | mc | pad)
  float* qh   = qi + 16 * 36;        // [16][64]  query hidden
  float* Ss   = qh + 16 * 64;        // [256][20] scores^T [led][q]
  float* dist = Ss + 256 * 20;       // [256]
  float* k1b  = dist + 256;          // per-wave [16][36]
  float* k2b  = k1b + 8 * 16 * 36;   // per-wave [16][132]
  __shared__ unsigned maskw[256];
  __shared__ float cws[16];
  __shared__ float s2i_sh;
  __shared__ float redA[256];
  __shared__ float redB[256];

  const int t = threadIdx.x;
  const int b = blockIdx.x;

  // ---- stage 0: stage weights / per-led data into LDS
  for (int i = t; i < 6 * 96; i += 256) {
    int p = i / 96, c = i % 96;
    float v = 0.f;
    if (c < 64) {
      int k = 2 * p + (c & 1), n = c >> 1;
      if (k < 11) v = kw1[k * 32 + n];
    }
    w1p[i] = v;
  }
  for (int i = t; i < 16 * 288; i += 256) {
    int p = i / 288, c = i % 288;
    w2p[i] = (c < 256) ? kw2[(2 * p + (c & 1)) * 128 + (c >> 1)] : 0.f;
  }
  for (int i = t; i < 16 * 36; i += 256) {
    int r = i / 36, c = i % 36;
    float v = 0.f;
    if (r < 12) {
      if (c == 0) v = rss_t[(size_t)b * 12 + r];
      else if (c < 33) v = mc_ws[(size_t)b * 32 + (c - 1)];
    }
    qi[i] = v;
  }
  {
    int led = t;
    const float* pp = prevp + (size_t)b * 3;
    const float* lp = ledp + ((size_t)b * 256 + led) * 3;
    float dx = pp[0] - lp[0], dy = pp[1] - lp[1], dz = pp[2] - lp[2];
    dist[led] = dx * dx + dy * dy + dz * dz;
    unsigned m = 0;
    for (int q = 0; q < 12; ++q) m |= (fmask[q * 256 + led] != 0 ? 1u : 0u) << q;
    maskw[led] = m;
  }
  if (t < 16) cws[t] = (t < 12) ? out[CW_OFF + (size_t)b * 12 + t] : 0.f;
  if (t == 0) {
    float s = sig_ws[b];
    s2i_sh = 1.f / (2.f * s * s);
  }
  __syncthreads();

  // ---- stage 1: qh = relu(qi @ qf_w1 + b)   (tiny; VALU, co-executes)
#pragma unroll
  for (int i = 0; i < 4; ++i) {
    int e = t + 256 * i, r = e >> 6, c = e & 63;
    float acc = qb1[c];
    for (int k = 0; k < 33; ++k) acc = fmaf(qi[r * 36 + k], qw1[k * 64 + c], acc);
    qh[r * 64 + c] = fmaxf(acc, 0.f);
  }
  __syncthreads();

  // ---- stage 2: Qs = (qh @ qf_w2 + b) * channel_weight
#pragma unroll
  for (int i = 0; i < 8; ++i) {
    int e = t + 256 * i, r = e >> 7, c = e & 127;
    float acc = qb2[c];
    for (int k = 0; k < 64; ++k) acc = fmaf(qh[r * 64 + k], qw2[k * 128 + c], acc);
    Qs[r * 132 + c] = acc * cws[r];
  }
  __syncthreads();

  // ---- stage 3: per-wave WMMA pipeline over two 16-LED tiles
  const int lane = t & 31;
  const int wv = t >> 5;
  const int halo = lane >> 4;   // 0: lanes 0-15, 1: lanes 16-31
  const int col = lane & 15;    // M (A) / N (B,C,D) index
  const float s2i = s2i_sh;
  float* kb = k1b + wv * (16 * 36);
  float* k2 = k2b + wv * (16 * 132);

#pragma unroll
  for (int mbi = 0; mbi < 2; ++mbi) {
    const int mb = 2 * wv + mbi;
    // --- K layer 1: [16x12] x [12x32] (K padded 11->12 with zeros)
    v8f c0 = splat8(kb1[col]);
    v8f c1 = splat8(kb1[16 + col]);
    const float* fea = ledf + ((size_t)b * 256 + mb * 16 + col) * 8;
    const float* pos = ledp + ((size_t)b * 256 + mb * 16 + col) * 3;
    const float p0 = pos[0], p1 = pos[1], p2 = pos[2];
#pragma unroll
    for (int ks = 0; ks < 3; ++ks) {
      v2f a;
      if (ks == 0)      { a.x = fea[2 * halo];     a.y = fea[2 * halo + 1]; }
      else if (ks == 1) { a.x = fea[4 + 2 * halo]; a.y = fea[5 + 2 * halo]; }
      else              { a.x = halo ? p2 : p0;    a.y = halo ? 0.f : p1; }
      int row0 = 4 * ks + 2 * halo;
      c0 = wmma4(a, ldsBp(w1p, 96, row0, col), c0);
      c1 = wmma4(a, ldsBp(w1p, 96, row0, 16 + col), c1);
    }
#pragma unroll
    for (int r = 0; r < 8; ++r) {
      int m = r + 8 * halo;
      kb[m * 36 + col] = fmaxf(c0[r], 0.f);
      kb[m * 36 + 16 + col] = fmaxf(c1[r], 0.f);
    }

    // --- K layer 2: [16x32] x [32x128]
    v8f acc[8];
#pragma unroll
    for (int nb = 0; nb < 8; ++nb) acc[nb] = splat8(kb2[nb * 16 + col]);
#pragma unroll
    for (int ks = 0; ks < 8; ++ks) {
      int row0 = 4 * ks + 2 * halo;
      v2f a;
      a.x = kb[col * 36 + row0];
      a.y = kb[col * 36 + row0 + 1];
#pragma unroll
      for (int nb = 0; nb < 8; ++nb)
        acc[nb] = wmma4(a, ldsBp(w2p, 288, row0, nb * 16 + col), acc[nb]);
    }
#pragma unroll
    for (int nb = 0; nb < 8; ++nb) {
#pragma unroll
      for (int r = 0; r < 8; ++r) {
        int m = r + 8 * halo;
        k2[m * 132 + nb * 16 + col] = acc[nb][r];
      }
    }

    // --- scores^T tile: [16 led x 128] x [128 x 16 q]  (B = Q^T from LDS)
    v8f sc = splat8(0.f);
#pragma unroll
    for (int ks = 0; ks < 32; ++ks) {
      int row0 = 4 * ks + 2 * halo;
      v2f a, bq;
      a.x = k2[col * 132 + row0];
      a.y = k2[col * 132 + row0 + 1];
      bq.x = Qs[col * 132 + row0];
      bq.y = Qs[col * 132 + row0 + 1];
      sc = wmma4(a, bq, sc);
    }
#pragma unroll
    for (int r = 0; r < 8; ++r) {
      int led = mb * 16 + r + 8 * halo;
      float v = sc[r] - dist[led] * s2i;
      if (!((maskw[led] >> col) & 1u)) v = -FLT_MAX;
      Ss[led * 20 + col] = v;
    }
  }
  __syncthreads();

  // ---- stage 4: softmax over 256 LEDs per (q) row
  {
    const int q = t >> 4;      // 0..15 (rows 12..15 are dummies)
    const int chunk = t & 15;  // 16 LEDs per thread
    float lmax = -FLT_MAX;
#pragma unroll
    for (int i = 0; i < 16; ++i)
      lmax = fmaxf(lmax, Ss[(chunk * 16 + i) * 20 + q]);
    redA[q * 16 + chunk] = lmax;
    __syncthreads();
    float rmax = -FLT_MAX;
#pragma unroll
    for (int j = 0; j < 16; ++j) rmax = fmaxf(rmax, redA[q * 16 + j]);
    float lsum = 0.f;
#pragma unroll
    for (int i = 0; i < 16; ++i) {
      int led = chunk * 16 + i;
      float e = expf(Ss[led * 20 + q] - rmax);
      Ss[led * 20 + q] = e;
      lsum += e;
    }
    redB[q * 16 + chunk] = lsum;
    __syncthreads();
    float rsum = 0.f;
#pragma unroll
    for (int j = 0; j < 16; ++j) rsum += redB[q * 16 + j];
    float inv = 1.f / rsum;
    if (q < 12) {
      float* op = out + (size_t)b * NRSS * NLED + (size_t)q * NLED + chunk * 16;
#pragma unroll
      for (int i = 0; i < 16; ++i) op[i] = Ss[(chunk * 16 + i) * 20 + q] * inv;
    }
  }
}

extern "C" void kernel_launch(void* const* d_in, const int* in_sizes, int n_in,
                              void* d_out, int out_size, void* d_ws,
                              size_t ws_size, hipStream_t stream) {
  (void)in_sizes; (void)n_in; (void)out_size; (void)ws_size;
  const float* rss   = (const float*)d_in[0];
  const float* ledf  = (const float*)d_in[1];
  const float* ledp  = (const float*)d_in[2];
  const float* prevp = (const float*)d_in[3];
  const int*   fmask = (const int*)d_in[4];
  const float* lstm  = (const float*)d_in[5];
  const float* me_w1 = (const float*)d_in[6];
  const float* me_b1 = (const float*)d_in[7];
  const float* me_w2 = (const float*)d_in[8];
  const float* me_b2 = (const float*)d_in[9];
  const float* ci_w  = (const float*)d_in[10];
  const float* ci_b  = (const float*)d_in[11];
  const float* qf_w1 = (const float*)d_in[12];
  const float* qf_b1 = (const float*)d_in[13];
  const float* qf_w2 = (const float*)d_in[14];
  const float* qf_b2 = (const float*)d_in[15];
  const float* kp_w1 = (const float*)d_in[16];
  const float* kp_b1 = (const float*)d_in[17];
  const float* kp_w2 = (const float*)d_in[18];
  const float* kp_b2 = (const float*)d_in[19];
  const float* ss_w  = (const float*)d_in[20];
  const float* ss_b  = (const float*)d_in[21];

  float* outp   = (float*)d_out;
  float* mc_ws  = (float*)d_ws;                // B*32 floats
  float* sig_ws = mc_ws + (size_t)NB * 32;     // B floats

  const size_t smem = (size_t)(6 * 96 + 16 * 288 + 16 * 132 + 16 * 36 +
                               16 * 64 + 256 * 20 + 256 + 8 * 16 * 36 +
                               8 * 16 * 132) * sizeof(float);  // ~143 KB
  (void)hipFuncSetAttribute((const void*)k2_attn,
                            hipFuncAttributeMaxDynamicSharedMemorySize,
                            (int)smem);

  k1_motion<<<NB / 16, 256, 0, stream>>>(lstm, me_w1, me_b1, me_w2, me_b2,
                                         ci_w, ci_b, ss_w, ss_b, outp, mc_ws,
                                         sig_ws);
  k2_attn<<<NB, 256, smem, stream>>>(rss, ledf, ledp, prevp, fmask, qf_w1,
                                     qf_b1, qf_w2, qf_b2, kp_w1, kp_b1, kp_w2,
                                     kp_b2, mc_ws, sig_ws, outp);
}